// StatefulRecurrent_33698313404692
// MI455X (gfx1250) — compile-verified
//
#include <hip/hip_runtime.h>

// Diagonal complex linear recurrence h[t] = A*h[t-1] + x[t], 3-phase chunked scan.
// Phase 1: per-chunk local scans -> chunk end states (reads x once).
// Phase 2: tiny serial prefix over chunk carries with A^CHUNK_T (6 squarings).
// Phase 3: re-scan each chunk from its true carry, stream (B,T,C,2) output.
//
// CDNA5 specifics: barrier-free double-buffered prefetch via
// global_load_async_to_lds_b128 (saddr form) + s_wait_asynccnt; each thread
// async-copies exactly the float4 lanes it consumes, so LDS is a wave-private
// staging buffer and no workgroup barriers are needed. Output uses
// non-temporal B128 stores to keep x resident in L2 between phases 1 and 3.

#define THREADS   128            // threads per block (4 wave32)
#define C_BLK     512            // channels per block (4 per thread)
#define STAGE_T   8              // timesteps per async stage
#define CHUNK_T   64             // timesteps per chunk
#define NSTAGES   (CHUNK_T / STAGE_T)

typedef float v4f __attribute__((ext_vector_type(4)));

#define WAIT_ASYNCCNT(N) asm volatile("s_wait_asynccnt " #N ::: "memory")

// complex step: (hr,hi) = A*(hr,hi) + (xv, 0)
#define CSTEP(HR, HI, AR, AI, XV)                                  \
    {                                                              \
        float _nr = fmaf((HR), (AR), fmaf(-(HI), (AI), (XV)));     \
        float _ni = fmaf((HR), (AI), (HI) * (AR));                 \
        (HR) = _nr; (HI) = _ni;                                    \
    }

// Issue one stage: STAGE_T rows x C_BLK floats, contiguous in global memory.
// Thread tid copies float4 column tid of each row -> exactly what it consumes.
__device__ __forceinline__ void issue_stage(const float* __restrict__ x,
                                            const float* lds_buf,   // smem[buf]
                                            unsigned gbase_bytes,   // byte off of (row t0s, this thread's float4)
                                            unsigned rowBytes,      // C*4
                                            int tid)
{
    // Ensure previously issued LDS reads of the buffer being overwritten have
    // fully retired before async writes can land (belt-and-braces; compiler
    // has already consumed them via the recurrence dependency chain).
    asm volatile("s_wait_dscnt 0x0" ::: "memory");
#pragma unroll
    for (int t = 0; t < STAGE_T; ++t) {
        unsigned goff = gbase_bytes + (unsigned)t * rowBytes;
        // flat(shared) low 32 bits == wave-relative LDS byte address (ISA 10.2)
        unsigned lds = (unsigned)(unsigned long long)(const void*)
                           (lds_buf + (t * C_BLK + tid * 4));
        asm volatile("global_load_async_to_lds_b128 %0, %1, %2"
                     :: "v"(lds), "v"(goff), "s"(x)
                     : "memory");
    }
}

// ---------------- Phase 1: per-chunk local scan (h starts at 0) ----------------
__global__ void __launch_bounds__(THREADS)
k_chunk_end(const float* __restrict__ x,
            const float* __restrict__ Ar_g, const float* __restrict__ Ai_g,
            float2* __restrict__ wsEnd,
            int B, int T, int C)
{
    __shared__ float smem[2][STAGE_T * C_BLK];

    const int tid = threadIdx.x;
    const int q   = blockIdx.x * THREADS + tid;   // float4 channel group
    const int b   = blockIdx.y;
    const int j   = blockIdx.z;
    const int t0  = j * CHUNK_T;
    const unsigned rowBytes = (unsigned)C * 4u;

    const float4 Ar = ((const float4*)Ar_g)[q];
    const float4 Ai = ((const float4*)Ai_g)[q];

    float hr0 = 0.f, hr1 = 0.f, hr2 = 0.f, hr3 = 0.f;
    float hi0 = 0.f, hi1 = 0.f, hi2 = 0.f, hi3 = 0.f;

    const unsigned gstage0 = ((unsigned)(b * T + t0) * (unsigned)C + (unsigned)q * 4u) * 4u;

    issue_stage(x, smem[0], gstage0, rowBytes, tid);

#pragma unroll
    for (int s = 0; s < NSTAGES; ++s) {
        if (s + 1 < NSTAGES) {
            issue_stage(x, smem[(s + 1) & 1],
                        gstage0 + (unsigned)(s + 1) * STAGE_T * rowBytes, rowBytes, tid);
            WAIT_ASYNCCNT(8);   // stage s complete; stage s+1 in flight
        } else {
            WAIT_ASYNCCNT(0);
        }
        const float4* buf = (const float4*)smem[s & 1];
#pragma unroll
        for (int t = 0; t < STAGE_T; ++t) {
            float4 xv = buf[t * (C_BLK / 4) + tid];   // this thread's own async'd data
            CSTEP(hr0, hi0, Ar.x, Ai.x, xv.x);
            CSTEP(hr1, hi1, Ar.y, Ai.y, xv.y);
            CSTEP(hr2, hi2, Ar.z, Ai.z, xv.z);
            CSTEP(hr3, hi3, Ar.w, Ai.w, xv.w);
        }
    }

    // chunk end state: float2 per channel at ((j*B+b)*C + c)
    float4* dst = (float4*)(wsEnd + ((long long)j * B + b) * C) + 2 * q;
    dst[0] = make_float4(hr0, hi0, hr1, hi1);
    dst[1] = make_float4(hr2, hi2, hr3, hi3);
}

// ---------------- Phase 2: serial prefix over chunk carries ----------------
__global__ void __launch_bounds__(256)
k_scan_chunks(const float* __restrict__ Ar_g, const float* __restrict__ Ai_g,
              const float* __restrict__ h0r_g, const float* __restrict__ h0i_g,
              const float2* __restrict__ wsEnd, float2* __restrict__ wsCarry,
              int B, int C, int J)
{
    long long idx = (long long)blockIdx.x * blockDim.x + threadIdx.x;
    if (idx >= (long long)B * C) return;
    const int b = (int)(idx / C);
    const int c = (int)(idx % C);

    const float Ar = Ar_g[c];
    const float Ai = Ai_g[c];

    // A^CHUNK_T via repeated squaring (CHUNK_T = 64 = 2^6)
    float pr = Ar, pi = Ai;
#pragma unroll
    for (int i = 0; i < 6; ++i) {
        float nr = pr * pr - pi * pi;
        float ni = 2.f * pr * pi;
        pr = nr; pi = ni;
    }

    float cr = h0r_g[idx];
    float ci = h0i_g[idx];
    for (int j = 0; j < J; ++j) {
        wsCarry[((long long)j * B + b) * C + c] = make_float2(cr, ci);
        float2 e = wsEnd[((long long)j * B + b) * C + c];
        float nr = fmaf(cr, pr, fmaf(-ci, pi, e.x));
        float ni = fmaf(cr, pi, fmaf(ci, pr, e.y));
        cr = nr; ci = ni;
    }
}

// ---------------- Phase 3: re-scan chunks from true carry, write output ----------------
__global__ void __launch_bounds__(THREADS)
k_chunk_out(const float* __restrict__ x,
            const float* __restrict__ Ar_g, const float* __restrict__ Ai_g,
            const float2* __restrict__ wsCarry,
            float* __restrict__ out,            // (B,T,C,2)
            int B, int T, int C)
{
    __shared__ float smem[2][STAGE_T * C_BLK];

    const int tid = threadIdx.x;
    const int q   = blockIdx.x * THREADS + tid;
    const int b   = blockIdx.y;
    const int j   = blockIdx.z;
    const int t0  = j * CHUNK_T;
    const unsigned rowBytes = (unsigned)C * 4u;

    const float4 Ar = ((const float4*)Ar_g)[q];
    const float4 Ai = ((const float4*)Ai_g)[q];

    // carry-in (true h[t0-1]) for this thread's 4 channels
    const float4* cin = (const float4*)(wsCarry + ((long long)j * B + b) * C) + 2 * q;
    float4 c01 = cin[0];
    float4 c23 = cin[1];
    float hr0 = c01.x, hi0 = c01.y, hr1 = c01.z, hi1 = c01.w;
    float hr2 = c23.x, hi2 = c23.y, hr3 = c23.z, hi3 = c23.w;

    const unsigned gstage0 = ((unsigned)(b * T + t0) * (unsigned)C + (unsigned)q * 4u) * 4u;

    issue_stage(x, smem[0], gstage0, rowBytes, tid);

#pragma unroll
    for (int s = 0; s < NSTAGES; ++s) {
        if (s + 1 < NSTAGES) {
            issue_stage(x, smem[(s + 1) & 1],
                        gstage0 + (unsigned)(s + 1) * STAGE_T * rowBytes, rowBytes, tid);
            WAIT_ASYNCCNT(8);
        } else {
            WAIT_ASYNCCNT(0);
        }
        const float4* buf = (const float4*)smem[s & 1];
#pragma unroll
        for (int t = 0; t < STAGE_T; ++t) {
            float4 xv = buf[t * (C_BLK / 4) + tid];
            CSTEP(hr0, hi0, Ar.x, Ai.x, xv.x);
            CSTEP(hr1, hi1, Ar.y, Ai.y, xv.y);
            CSTEP(hr2, hi2, Ar.z, Ai.z, xv.z);
            CSTEP(hr3, hi3, Ar.w, Ai.w, xv.w);

            // out[((b*T+t)*C + c)*2 + {0,1}] : 32 contiguous bytes per thread
            long long o2 = ((long long)b * T + (t0 + s * STAGE_T + t)) * C + (long long)q * 4;
            v4f lo; lo.x = hr0; lo.y = hi0; lo.z = hr1; lo.w = hi1;
            v4f hi; hi.x = hr2; hi.y = hi2; hi.z = hr3; hi.w = hi3;
            __builtin_nontemporal_store(lo, (v4f*)(out + o2 * 2));
            __builtin_nontemporal_store(hi, (v4f*)(out + o2 * 2 + 4));
        }
    }
}

extern "C" void kernel_launch(void* const* d_in, const int* in_sizes, int n_in,
                              void* d_out, int out_size, void* d_ws, size_t ws_size,
                              hipStream_t stream)
{
    (void)n_in; (void)out_size; (void)ws_size;

    const float* x   = (const float*)d_in[0];
    const float* Ar  = (const float*)d_in[1];
    const float* Ai  = (const float*)d_in[2];
    const float* h0r = (const float*)d_in[3];
    const float* h0i = (const float*)d_in[4];

    const int C = in_sizes[1];                                         // 512
    const int B = in_sizes[3] / C;                                     // 8
    const int T = (int)((long long)in_sizes[0] / ((long long)B * C));  // 4096
    const int J = T / CHUNK_T;                                         // 64

    float2* wsEnd   = (float2*)d_ws;                 // J*B*C float2 (2 MB)
    float2* wsCarry = wsEnd + (size_t)J * B * C;     // J*B*C float2 (2 MB)

    dim3 blk(THREADS);
    dim3 g1(C / C_BLK, B, J);

    hipLaunchKernelGGL(k_chunk_end, g1, blk, 0, stream, x, Ar, Ai, wsEnd, B, T, C);

    int nbc = B * C;
    hipLaunchKernelGGL(k_scan_chunks, dim3((nbc + 255) / 256), dim3(256), 0, stream,
                       Ar, Ai, h0r, h0i, wsEnd, wsCarry, B, C, J);

    hipLaunchKernelGGL(k_chunk_out, g1, blk, 0, stream, x, Ar, Ai, wsCarry,
                       (float*)d_out, B, T, C);
}